// FernBitWord_37512244364033
// MI455X (gfx1250) — compile-verified
//
#include <hip/hip_runtime.h>

// Problem constants (from the reference)
#define PADV   3
#define NN     32
#define HH     160
#define WW     160
#define NF     128          // 16*8 fern features

// Tiling
#define TH      16              // output rows per block tile
#define LROWS   (TH + 6)        // padded Tp rows needed: y1 in [0,5] plus +1 bilinear
#define LSTRIDE 168             // LDS row stride in floats (166 cols + pad)
#define FSPLIT  4               // split feature dim across blocks for occupancy
#define FPB     (NF / FSPLIT)   // features per block

// ---------------------------------------------------------------------------
// Prep kernel: one block of 128 threads.
//  - L = max(|dx1|,|dx2|,|dy1|,|dy2|) over all 128 features
//  - per-feature window start offsets (pre-flattened into LDS element offsets)
//    with jax.lax.dynamic_slice clamping to [0, 2*PAD-1] and C truncation
//    matching jnp astype(int32)
//  - fractional bilinear weights + thresholds
// ws layout (4B units): [0:128) offA(int) | [128:256) offB(int) |
//   [256:384) fxa | [384:512) fya | [512:640) fxb | [640:768) fyb | [768:896) th
// ---------------------------------------------------------------------------
__global__ __launch_bounds__(128) void fern_prep(
    const float* __restrict__ dx1, const float* __restrict__ dx2,
    const float* __restrict__ dy1, const float* __restrict__ dy2,
    const float* __restrict__ th,  float* __restrict__ ws)
{
    __shared__ float red[128];
    const int t = threadIdx.x;
    const float a1 = dx1[t], a2 = dx2[t], b1 = dy1[t], b2 = dy2[t];
    red[t] = fmaxf(fmaxf(fabsf(a1), fabsf(a2)), fmaxf(fabsf(b1), fabsf(b2)));
    __syncthreads();
    for (int s = 64; s >= 1; s >>= 1) {
        if (t < s) red[t] = fmaxf(red[t], red[t + s]);
        __syncthreads();
    }
    const float L = red[0];
    const int shift = PADV - (int)ceilf(L);

    int x1a = (int)(L + floorf(a1)) + shift;   // trunc-toward-zero == astype(int32)
    int y1a = (int)(L + floorf(b1)) + shift;
    int x1b = (int)(L + floorf(a2)) + shift;
    int y1b = (int)(L + floorf(b2)) + shift;
    // dynamic_slice clamps start to [0, (W+2P)-(W+1)] = [0, 5]
    x1a = min(max(x1a, 0), 2 * PADV - 1);
    y1a = min(max(y1a, 0), 2 * PADV - 1);
    x1b = min(max(x1b, 0), 2 * PADV - 1);
    y1b = min(max(y1b, 0), 2 * PADV - 1);

    int* offA = (int*)ws;
    int* offB = offA + NF;
    float* fq = ws + 2 * NF;
    offA[t] = y1a * LSTRIDE + x1a;
    offB[t] = y1b * LSTRIDE + x1b;
    fq[0 * NF + t] = a1 - floorf(a1);   // fxa
    fq[1 * NF + t] = b1 - floorf(b1);   // fya
    fq[2 * NF + t] = a2 - floorf(a2);   // fxb
    fq[3 * NF + t] = b2 - floorf(b2);   // fyb
    fq[4 * NF + t] = th[t];
}

// ---------------------------------------------------------------------------
// Main kernel: grid (32 images, 10 row-tiles, 4 feature-splits), 256 threads.
// Stage padded 22x166 input tile into LDS via CDNA5 async global->LDS DMA,
// then each wave owns a row-pair; each lane computes 2 vertically adjacent
// pixels per (feature, col-chunk) sharing the middle interp row.
// ---------------------------------------------------------------------------
__global__ __launch_bounds__(256) void fern_main(
    const float* __restrict__ x, const float* __restrict__ ws,
    float* __restrict__ out)
{
    __shared__ float tile[LROWS * LSTRIDE];   // 22*168*4 = 14784 B

    const int n   = blockIdx.x;
    const int h0  = blockIdx.y * TH;
    const int f0  = blockIdx.z * FPB;
    const int tid = threadIdx.x;

    // Zero LDS tile (provides the zero padding of Tp)
    for (int i = tid; i < LROWS * LSTRIDE; i += 256) tile[i] = 0.0f;
    __syncthreads();

    // Async-load in-bounds elements of Tp[h0 .. h0+LROWS) x [0,166) into LDS.
    const float* chan = x + ((size_t)n * 3 + 1) * (HH * WW);   // x[n,1,:,:]
    const unsigned ldsBase = (unsigned)(unsigned long long)(void*)tile; // LDS byte offset
    for (int i = tid; i < LROWS * 166; i += 256) {
        const int lr = i / 166;
        const int c  = i - lr * 166;
        const int gr = h0 + lr - PADV;
        const int gc = c - PADV;
        if (gr >= 0 && gr < HH && gc >= 0 && gc < WW) {
            const unsigned ldsOff = ldsBase + 4u * (unsigned)(lr * LSTRIDE + c);
            const float* gp = chan + gr * WW + gc;
#if defined(__HIP_DEVICE_COMPILE__)
            asm volatile("global_load_async_to_lds_b32 %0, %1, off"
                         :: "v"(ldsOff), "v"(gp) : "memory");
#else
            (void)ldsOff; (void)gp;
#endif
        }
    }
#if defined(__HIP_DEVICE_COMPILE__)
    asm volatile("s_wait_asynccnt 0" ::: "memory");
#endif
    __syncthreads();

    // Per-feature parameter tables (uniform indices -> scalar loads)
    const int*   offA = (const int*)ws;
    const int*   offB = offA + NF;
    const float* fxa_t = (const float*)ws + 2 * NF;
    const float* fya_t = fxa_t + NF;
    const float* fxb_t = fxa_t + 2 * NF;
    const float* fyb_t = fxa_t + 3 * NF;
    const float* thv_t = fxa_t + 4 * NF;

    const int lane = tid & 31;
    const int rl0  = (tid >> 5) * 2;           // local row pair: rl0, rl0+1
    float* outN = out + (size_t)n * NF * HH * WW;

    for (int f = f0; f < f0 + FPB; ++f) {
        const int   oa  = offA[f], ob = offB[f];
        const float fxa = fxa_t[f], fya = fya_t[f];
        const float fxb = fxb_t[f], fyb = fyb_t[f];
        const float tv  = thv_t[f];

        const float* ta = &tile[oa + rl0 * LSTRIDE];
        const float* tb = &tile[ob + rl0 * LSTRIDE];
        float* orow = outN + ((size_t)f * HH + (h0 + rl0)) * WW;

        #pragma unroll
        for (int cch = 0; cch < 5; ++cch) {
            const int w = lane + 32 * cch;
            // Window A: 3 rows x 2 cols (middle row shared by both pixels)
            const float a00 = ta[w],               a01 = ta[w + 1];
            const float a10 = ta[LSTRIDE + w],     a11 = ta[LSTRIDE + w + 1];
            const float a20 = ta[2 * LSTRIDE + w], a21 = ta[2 * LSTRIDE + w + 1];
            // Window B
            const float b00 = tb[w],               b01 = tb[w + 1];
            const float b10 = tb[LSTRIDE + w],     b11 = tb[LSTRIDE + w + 1];
            const float b20 = tb[2 * LSTRIDE + w], b21 = tb[2 * LSTRIDE + w + 1];

            const float at0 = a00 + fxa * (a01 - a00);
            const float at1 = a10 + fxa * (a11 - a10);
            const float at2 = a20 + fxa * (a21 - a20);
            const float bt0 = b00 + fxb * (b01 - b00);
            const float bt1 = b10 + fxb * (b11 - b10);
            const float bt2 = b20 + fxb * (b21 - b20);

            const float ia0 = at0 + fya * (at1 - at0);
            const float ia1 = at1 + fya * (at2 - at1);
            const float ib0 = bt0 + fyb * (bt1 - bt0);
            const float ib1 = bt1 + fyb * (bt2 - bt1);

            const float v0 = fminf(fmaxf((ia0 - ib0 - tv + 0.5f) * 0.25f, 0.0f), 1.0f);
            const float v1 = fminf(fmaxf((ia1 - ib1 - tv + 0.5f) * 0.25f, 0.0f), 1.0f);

            // Streaming (non-temporal) stores: output is never re-read.
            __builtin_nontemporal_store(v0, &orow[w]);
            __builtin_nontemporal_store(v1, &orow[WW + w]);
        }
    }
}

extern "C" void kernel_launch(void* const* d_in, const int* in_sizes, int n_in,
                              void* d_out, int out_size, void* d_ws, size_t ws_size,
                              hipStream_t stream) {
    const float* x   = (const float*)d_in[0];
    const float* dx1 = (const float*)d_in[1];
    const float* dx2 = (const float*)d_in[2];
    const float* dy1 = (const float*)d_in[3];
    const float* dy2 = (const float*)d_in[4];
    const float* th  = (const float*)d_in[5];
    float* out = (float*)d_out;
    float* ws  = (float*)d_ws;   // needs 896 * 4 = 3584 bytes

    fern_prep<<<1, 128, 0, stream>>>(dx1, dx2, dy1, dy2, th, ws);
    dim3 grid(NN, HH / TH, FSPLIT);
    fern_main<<<grid, 256, 0, stream>>>(x, ws, out);
}